// DL_SOTA_PrototypeNet_57836029608242
// MI455X (gfx1250) — compile-verified
//
#include <hip/hip_runtime.h>
#include <math.h>

#define B_ 32
#define N_ 8192
#define P_ 128
#define H_ 64
#define D_ 256
#define K_ 6
#define TEMP_INV 10.0f
#define LN_EPS 1e-5f
#define WAVES 8
#define HROW 68   // padded h-row stride (floats) -> conflict-free frag loads

// workspace layout (floats)
#define WS_ACC   0          // 2*32*6 = 384 accumulators (count, varnum)
#define WS_W2P   512        // 64x16 folded weights (cols 0-5: w2·p_k, col 6: w2·b2)
#define WS_G     1536       // 64x64 Gram matrix w2·w2^T
#define WS_SC    5632       // [0..5]=b2·p_k, [6]=|b2|^2

typedef float v2f __attribute__((ext_vector_type(2)));
typedef float v4f __attribute__((ext_vector_type(4)));
typedef float v8f __attribute__((ext_vector_type(8)));

__device__ __forceinline__ float rsum16(float v) {
  v += __shfl_xor(v, 1, 32);
  v += __shfl_xor(v, 2, 32);
  v += __shfl_xor(v, 4, 32);
  v += __shfl_xor(v, 8, 32);
  return v;
}

__device__ __forceinline__ float gelu_exact(float v) {
  return 0.5f * v * (1.0f + erff(v * 0.70710678118654752f));
}

// ---- one-block prep: fold prototypes/bias into the second-layer weights ----
__global__ __launch_bounds__(256) void protonet_prep(
    const float* __restrict__ w2, const float* __restrict__ b2,
    const float* __restrict__ proto, float* __restrict__ ws)
{
  float* W2P = ws + WS_W2P;
  float* G   = ws + WS_G;
  float* SC  = ws + WS_SC;
  const int t = threadIdx.x;

  for (int e = t; e < H_ * 16; e += 256) {
    int h = e >> 4, k = e & 15;
    float s = 0.f;
    if (k < K_) {
      for (int d = 0; d < D_; ++d) s += w2[h * D_ + d] * proto[k * D_ + d];
    } else if (k == K_) {
      for (int d = 0; d < D_; ++d) s += w2[h * D_ + d] * b2[d];
    }
    W2P[e] = s;
  }
  for (int e = t; e < H_ * H_; e += 256) {
    int i = e >> 6, j = e & 63;
    float s = 0.f;
    for (int d = 0; d < D_; ++d) s += w2[i * D_ + d] * w2[j * D_ + d];
    G[e] = s;
  }
  if (t < K_) {
    float s = 0.f;
    for (int d = 0; d < D_; ++d) s += b2[d] * proto[t * D_ + d];
    SC[t] = s;
  }
  if (t == K_) {
    float s = 0.f;
    for (int d = 0; d < D_; ++d) s += b2[d] * b2[d];
    SC[K_] = s;
  }
}

__global__ __launch_bounds__(256) void protonet_main(
    const float* __restrict__ x,  const float* __restrict__ w1,
    const float* __restrict__ b1, const float* __restrict__ lng,
    const float* __restrict__ lnb, const float* __restrict__ ws,
    float* __restrict__ acc)
{
  // k-quad interleaved layouts: elem (k,n) at [(k>>2)*NCOLS + n]*4 + (k&3)
  __shared__ float w1q[(P_ / 4) * H_ * 4];   // 8192 floats
  __shared__ float Gq[(H_ / 4) * H_ * 4];    // 4096 floats
  __shared__ float Pq[(H_ / 4) * 16 * 4];    // 1024 floats
  __shared__ float b1_s[H_], lng_s[H_], lnb_s[H_];
  __shared__ float h_s[WAVES * 16 * HROW];   // 8704 floats (~87 KB total LDS)

  const float* W2P = ws + WS_W2P;
  const float* G   = ws + WS_G;
  const float* SC  = ws + WS_SC;

  const int tid = threadIdx.x;
  for (int e = tid; e < P_ * H_; e += 256) {
    int k = e >> 6, n = e & 63;
    w1q[((k >> 2) * H_ + n) * 4 + (k & 3)] = w1[e];
  }
  for (int e = tid; e < H_ * H_; e += 256) {
    int k = e >> 6, n = e & 63;
    Gq[((k >> 2) * H_ + n) * 4 + (k & 3)] = G[e];
  }
  for (int e = tid; e < H_ * 16; e += 256) {
    int k = e >> 4, n = e & 15;
    Pq[((k >> 2) * 16 + n) * 4 + (k & 3)] = W2P[e];
  }
  if (tid < H_) { b1_s[tid] = b1[tid]; lng_s[tid] = lng[tid]; lnb_s[tid] = lnb[tid]; }
  __syncthreads();

  const int lane = tid & 31;
  const int wave = tid >> 5;
  const int s    = lane & 15;   // M index (A) / N sub-index (B,C)
  const int hi   = lane >> 4;   // K-pair selector
  const int row0 = (blockIdx.x * WAVES + wave) * 16;
  const int b_idx = row0 >> 13; // row0 / 8192

  // ---------------- GEMM1: h[16x64] = x[16x128] @ w1[128x64] ----------------
  v2f a1[32];
  const float* xrow = x + (size_t)(row0 + s) * P_;
  #pragma unroll
  for (int ks = 0; ks < 32; ++ks)
    a1[ks] = *(const v2f*)(xrow + 4 * ks + 2 * hi);

  v8f c1[4] = {};
  #pragma unroll
  for (int ks = 0; ks < 32; ++ks) {
    #pragma unroll
    for (int nt = 0; nt < 4; ++nt) {
      v2f bf = *(const v2f*)(&w1q[(ks * H_ + 16 * nt + s) * 4 + 2 * hi]);
      c1[nt] = __builtin_amdgcn_wmma_f32_16x16x4_f32(
          false, a1[ks], false, bf, (short)0, c1[nt], false, false);
    }
  }

  // bias + exact GELU; scatter C layout (lane: n=16*nt+s, vgpr r: m=8*hi+r) to LDS
  float* hw = &h_s[wave * 16 * HROW];
  #pragma unroll
  for (int nt = 0; nt < 4; ++nt) {
    int n = 16 * nt + s;
    float bv = b1_s[n];
    #pragma unroll
    for (int r = 0; r < 8; ++r) {
      float v = c1[nt][r] + bv;
      hw[(8 * hi + r) * HROW + n] = gelu_exact(v);
    }
  }
  asm volatile("s_wait_dscnt 0x0" ::: "memory");

  // ---------------- LayerNorm over H=64, in place ----------------
  {
    v4f vv[16];
    const v4f* rp = (const v4f*)(hw + s * HROW);
    #pragma unroll
    for (int i = 0; i < 16; ++i) vv[i] = rp[i];
    float sum = 0.f;
    #pragma unroll
    for (int i = 0; i < 16; ++i) sum += vv[i][0] + vv[i][1] + vv[i][2] + vv[i][3];
    float mu = sum * (1.0f / 64.0f);
    float var = 0.f;
    #pragma unroll
    for (int i = 0; i < 16; ++i) {
      float d0 = vv[i][0] - mu, d1 = vv[i][1] - mu, d2 = vv[i][2] - mu, d3 = vv[i][3] - mu;
      var += d0 * d0 + d1 * d1 + d2 * d2 + d3 * d3;
    }
    float rstd = rsqrtf(var * (1.0f / 64.0f) + LN_EPS);
    #pragma unroll
    for (int j = 0; j < 32; ++j) {          // lanes (s, s+16) each write one half
      float e0 = vv[j >> 2][j & 3];
      float e1 = vv[8 + (j >> 2)][j & 3];
      float e  = hi ? e1 : e0;
      int   k  = hi * 32 + j;
      hw[s * HROW + k] = (e - mu) * rstd * lng_s[k] + lnb_s[k];
    }
  }
  asm volatile("s_wait_dscnt 0x0" ::: "memory");

  // A fragments of normalized h (shared by both remaining GEMMs)
  v2f a2[16];
  #pragma unroll
  for (int ks = 0; ks < 16; ++ks)
    a2[ks] = *(const v2f*)(hw + s * HROW + 4 * ks + 2 * hi);

  // ---- GEMM_P: [16x16] = hn[16x64] @ W2P[64x16]  (cols 0-5 logits', col 6 = hn·u) ----
  v8f cp = {};
  #pragma unroll
  for (int ks = 0; ks < 16; ++ks) {
    v2f bf = *(const v2f*)(&Pq[(ks * 16 + s) * 4 + 2 * hi]);
    cp = __builtin_amdgcn_wmma_f32_16x16x4_f32(
        false, a2[ks], false, bf, (short)0, cp, false, false);
  }

  // ---- GEMM_G streamed: T = hn @ G, then z2 row-dot  z2[m] = sum_n T[m][n]*hn[m][n] ----
  float z2p[8];
  #pragma unroll
  for (int r = 0; r < 8; ++r) z2p[r] = 0.f;

  for (int nt = 0; nt < 4; ++nt) {
    v8f cg = {};
    #pragma unroll
    for (int ks = 0; ks < 16; ++ks) {
      v2f bf = *(const v2f*)(&Gq[(ks * H_ + 16 * nt + s) * 4 + 2 * hi]);
      cg = __builtin_amdgcn_wmma_f32_16x16x4_f32(
          false, a2[ks], false, bf, (short)0, cg, false, false);
    }
    int n = 16 * nt + s;
    #pragma unroll
    for (int r = 0; r < 8; ++r) {
      float hval = hw[(8 * hi + r) * HROW + n];
      z2p[r] += cg[r] * hval;
    }
  }

  // ---------------- per-row softmax + accumulation ----------------
  float b2p[K_];
  #pragma unroll
  for (int k = 0; k < K_; ++k) b2p[k] = SC[k];
  const float bb = SC[K_];

  float cnt[K_]  = {0, 0, 0, 0, 0, 0};
  float vnum[K_] = {0, 0, 0, 0, 0, 0};
  #pragma unroll
  for (int r = 0; r < 8; ++r) {
    float l[K_];
    #pragma unroll
    for (int k = 0; k < K_; ++k) l[k] = __shfl(cp[r], k, 16) + b2p[k];
    float dd = __shfl(cp[r], K_, 16);            // hn·(w2·b2)
    float zz = rsum16(z2p[r]) + 2.0f * dd + bb;  // |z|^2
    float mx = l[0] * TEMP_INV;
    #pragma unroll
    for (int k = 1; k < K_; ++k) mx = fmaxf(mx, l[k] * TEMP_INV);
    float e[K_];
    float es = 0.f;
    #pragma unroll
    for (int k = 0; k < K_; ++k) { e[k] = __expf(l[k] * TEMP_INV - mx); es += e[k]; }
    float inv = 1.0f / es;
    #pragma unroll
    for (int k = 0; k < K_; ++k) {
      float ak = e[k] * inv;
      cnt[k]  += ak;
      vnum[k] += ak * (zz - 2.0f * l[k]);        // |p_k|^2 term added in finalize
    }
  }
  #pragma unroll
  for (int k = 0; k < K_; ++k) {
    cnt[k]  += __shfl_xor(cnt[k], 16, 32);
    vnum[k] += __shfl_xor(vnum[k], 16, 32);
  }
  if (lane == 0) {
    #pragma unroll
    for (int k = 0; k < K_; ++k) {
      atomicAdd(&acc[b_idx * K_ + k], cnt[k]);
      atomicAdd(&acc[B_ * K_ + b_idx * K_ + k], vnum[k]);
    }
  }
}

__global__ void protonet_finalize(const float* __restrict__ proto,
                                  const float* __restrict__ acc,
                                  float* __restrict__ out)
{
  int t = threadIdx.x;
  if (t >= B_ * K_) return;
  int k = t % K_;
  float p2 = 0.f;
  for (int d = 0; d < D_; ++d) {
    float p = proto[k * D_ + d];
    p2 += p * p;
  }
  float c  = acc[t];
  float vn = acc[B_ * K_ + t];
  out[t] = (vn + p2 * c) / (c + 1e-6f);
}

extern "C" void kernel_launch(void* const* d_in, const int* in_sizes, int n_in,
                              void* d_out, int out_size, void* d_ws, size_t ws_size,
                              hipStream_t stream) {
  const float* x     = (const float*)d_in[0];
  const float* w1    = (const float*)d_in[1];
  const float* b1    = (const float*)d_in[2];
  const float* lng   = (const float*)d_in[3];
  const float* lnb   = (const float*)d_in[4];
  const float* w2    = (const float*)d_in[5];
  const float* b2    = (const float*)d_in[6];
  const float* proto = (const float*)d_in[7];
  float* ws = (float*)d_ws;

  hipMemsetAsync(ws + WS_ACC, 0, 2 * B_ * K_ * sizeof(float), stream);
  protonet_prep<<<dim3(1), dim3(256), 0, stream>>>(w2, b2, proto, ws);

  const int tiles = (B_ * N_) / (WAVES * 16);  // 2048 blocks, 1 wave per 16-row tile
  protonet_main<<<dim3(tiles), dim3(256), 0, stream>>>(x, w1, b1, lng, lnb, ws, ws + WS_ACC);
  protonet_finalize<<<dim3(1), dim3(256), 0, stream>>>(proto, ws + WS_ACC, (float*)d_out);
}